// ThalamicRNN_9560597201160
// MI455X (gfx1250) — compile-verified
//
#include <hip/hip_runtime.h>
#include <hip/hip_bf16.h>

typedef float v2f __attribute__((ext_vector_type(2)));
typedef float v8f __attribute__((ext_vector_type(8)));

#define NN    4096
#define NBG   64
#define BATCH 512

__device__ __forceinline__ float softplusf(float v) {
    // stable: logaddexp(v, 0) = max(v,0) + log1p(exp(-|v|))
    return fmaxf(v, 0.0f) + log1pf(expf(-fabsf(v)));
}

// -------------------------------------------------------------------------
// Kernel 1: t = r_thalamic ⊙ (V @ r)      (NBG x BATCH), t -> workspace
// grid.x = BATCH/64 = 8 blocks, 256 threads = 8 waves.
// Each wave computes a 16x32 strip of the 64x64 block (2 WMMA tiles).
// Double-buffered LDS, pair-major B so fragments are single ds_load_b64.
// -------------------------------------------------------------------------
__global__ __launch_bounds__(256)
void thal_vr_gate_kernel(const float* __restrict__ V,    // NBG x NN
                         const float* __restrict__ r,    // NN  x BATCH
                         const float* __restrict__ rth,  // NBG x BATCH
                         float* __restrict__ t)          // NBG x BATCH
{
    __shared__ float  ldsA[2][64][20];  // V tile: 64(M) x 16(K); stride 20: aligned pairs, no bank conflicts
    __shared__ float2 ldsB[2][8][80];   // r tile pair-major: [k/2][n] = {B[k,n], B[k+1,n]}

    const int tid  = threadIdx.x;
    const int wave = tid >> 5;
    const int lane = tid & 31;
    const int half = lane >> 4;
    const int l16  = lane & 15;

    const int nBase = blockIdx.x * 64;
    const int waveM = (wave & 3) * 16;    // 0,16,32,48
    const int waveN = (wave >> 2) * 32;   // 0 or 32

    v8f acc[2];
#pragma unroll
    for (int i = 0; i < 2; ++i)
#pragma unroll
        for (int j = 0; j < 8; ++j) acc[i][j] = 0.0f;

    auto stage = [&](int sidx, int buf) {
        const int kb = sidx * 16;
        // A tile: 64x16 floats, one float4 per thread (16B-aligned LDS rows)
        {
            const int row = tid >> 2, c4 = tid & 3;
            *reinterpret_cast<float4*>(&ldsA[buf][row][c4 * 4]) =
                *reinterpret_cast<const float4*>(&V[row * NN + kb + c4 * 4]);
        }
        // B tile pair-major: thread -> (n, k2); 4 coalesced b32 loads, 2 b64 stores
        {
            const int n = tid & 63, k2b = tid >> 6;   // k2b in 0..3
#pragma unroll
            for (int i = 0; i < 2; ++i) {
                const int k2 = k2b + i * 4;
                const float lo = r[(kb + 2 * k2 + 0) * BATCH + nBase + n];
                const float hi = r[(kb + 2 * k2 + 1) * BATCH + nBase + n];
                ldsB[buf][k2][n] = make_float2(lo, hi);
            }
        }
    };

    stage(0, 0);
    __syncthreads();

    for (int sidx = 0; sidx < NN / 16; ++sidx) {
        const int buf = sidx & 1;
        if (sidx + 1 < NN / 16) stage(sidx + 1, buf ^ 1);

#pragma unroll
        for (int s = 0; s < 4; ++s) {               // 4 x (K=4) sub-steps
            const v2f a = *reinterpret_cast<const v2f*>(
                &ldsA[buf][waveM + l16][s * 4 + 2 * half]);     // A[m, 4s+2h+{0,1}]
#pragma unroll
            for (int nt = 0; nt < 2; ++nt) {
                const v2f b = *reinterpret_cast<const v2f*>(
                    &ldsB[buf][s * 2 + half][waveN + nt * 16 + l16]);  // B[4s+2h+{0,1}, n]
                acc[nt] = __builtin_amdgcn_wmma_f32_16x16x4_f32(
                    false, a, false, b, (short)0, acc[nt], false, false);
            }
        }
        __syncthreads();
    }

    // epilogue: gate by r_thalamic, write t
#pragma unroll
    for (int nt = 0; nt < 2; ++nt) {
#pragma unroll
        for (int v = 0; v < 8; ++v) {
            const int row = waveM + v + 8 * half;              // C/D layout: M = v + 8h
            const int col = nBase + waveN + nt * 16 + l16;
            t[row * BATCH + col] = rth[row * BATCH + col] * acc[nt][v];
        }
    }
}

// -------------------------------------------------------------------------
// Kernel 2: rec = J @ r + U @ t ; x_new = x + (DT/TAU)(-x + rec + B);
//           r_new = softplus(x_new)
// grid = (NN/128, BATCH/64), 256 threads = 8 waves.
// Each wave owns a 16(M) x 64(N) strip: 4 WMMA tiles, 32 acc VGPRs.
// Concatenated K loop: stages 0..255 = (J, r), stages 256..259 = (U, t).
// J (64MB) < L2 (192MB), so re-reads across the 8 N-blocks hit L2; HBM
// traffic stays ~96MB total (~4us at 23.3 TB/s).
// -------------------------------------------------------------------------
__global__ __launch_bounds__(256)
void thal_rnn_step_kernel(const float* __restrict__ J,    // NN x NN
                          const float* __restrict__ U,    // NN x NBG
                          const float* __restrict__ t,    // NBG x BATCH (ws)
                          const float* __restrict__ r,    // NN x BATCH
                          const float* __restrict__ x,    // NN x BATCH
                          const float* __restrict__ Bv,   // NN x 1
                          float* __restrict__ out)        // [x_new | r_new]
{
    __shared__ float  ldsA[2][128][20]; // A tile: 128(M) x 16(K)
    __shared__ float2 ldsB[2][8][80];   // B tile pair-major: [k/2][n]

    const int tid  = threadIdx.x;
    const int wave = tid >> 5;
    const int lane = tid & 31;
    const int half = lane >> 4;
    const int l16  = lane & 15;

    const int mBase = blockIdx.x * 128;
    const int nBase = blockIdx.y * 64;

    v8f acc[4];
#pragma unroll
    for (int i = 0; i < 4; ++i)
#pragma unroll
        for (int j = 0; j < 8; ++j) acc[i][j] = 0.0f;

    constexpr int NSTAGES = NN / 16 + NBG / 16;   // 256 (J,r) + 4 (U,t)

    auto stage = [&](int sidx, int buf) {
        const float* __restrict__ Ap;
        const float* __restrict__ Bp;
        int lda, kb;
        if (sidx < NN / 16) { Ap = J; Bp = r; lda = NN;  kb = sidx * 16; }
        else                { Ap = U; Bp = t; lda = NBG; kb = (sidx - NN / 16) * 16; }
        // A tile: 128x16 floats = 512 float4, 2 per thread
#pragma unroll
        for (int i = 0; i < 2; ++i) {
            const int idx = tid + i * 256;
            const int row = idx >> 2, c4 = idx & 3;
            *reinterpret_cast<float4*>(&ldsA[buf][row][c4 * 4]) =
                *reinterpret_cast<const float4*>(&Ap[(mBase + row) * lda + kb + c4 * 4]);
        }
        // B tile pair-major: thread -> (n, k2); coalesced b32 loads
        {
            const int n = tid & 63, k2b = tid >> 6;
#pragma unroll
            for (int i = 0; i < 2; ++i) {
                const int k2 = k2b + i * 4;
                const float lo = Bp[(kb + 2 * k2 + 0) * BATCH + nBase + n];
                const float hi = Bp[(kb + 2 * k2 + 1) * BATCH + nBase + n];
                ldsB[buf][k2][n] = make_float2(lo, hi);
            }
        }
    };

    stage(0, 0);
    __syncthreads();

    for (int sidx = 0; sidx < NSTAGES; ++sidx) {
        const int buf = sidx & 1;
        if (sidx + 1 < NSTAGES) stage(sidx + 1, buf ^ 1);

#pragma unroll
        for (int s = 0; s < 4; ++s) {
            const v2f a = *reinterpret_cast<const v2f*>(
                &ldsA[buf][wave * 16 + l16][s * 4 + 2 * half]);
#pragma unroll
            for (int nt = 0; nt < 4; ++nt) {
                const v2f b = *reinterpret_cast<const v2f*>(
                    &ldsB[buf][s * 2 + half][nt * 16 + l16]);
                acc[nt] = __builtin_amdgcn_wmma_f32_16x16x4_f32(
                    false, a, false, b, (short)0, acc[nt], false, false);
            }
        }
        __syncthreads();
    }

    // epilogue: Euler update + softplus, write x_new and r_new
    const float dt_tau = 0.05f / 0.15f;
#pragma unroll
    for (int nt = 0; nt < 4; ++nt) {
#pragma unroll
        for (int v = 0; v < 8; ++v) {
            const int row = mBase + wave * 16 + v + 8 * half;  // C/D layout: M = v + 8h
            const int col = nBase + nt * 16 + l16;
            const float rec = acc[nt][v];
            const float xv  = x[row * BATCH + col];
            const float xn  = xv + dt_tau * (-xv + rec + Bv[row]);
            out[row * BATCH + col]              = xn;              // x_new
            out[NN * BATCH + row * BATCH + col] = softplusf(xn);   // r_new
        }
    }
}

extern "C" void kernel_launch(void* const* d_in, const int* in_sizes, int n_in,
                              void* d_out, int out_size, void* d_ws, size_t ws_size,
                              hipStream_t stream) {
    const float* rth = (const float*)d_in[0];  // (NBG, BATCH)
    const float* x   = (const float*)d_in[1];  // (NN, BATCH)
    const float* r   = (const float*)d_in[2];  // (NN, BATCH)
    const float* J   = (const float*)d_in[3];  // (NN, NN)
    const float* Bv  = (const float*)d_in[4];  // (NN, 1)
    const float* U   = (const float*)d_in[5];  // (NN, NBG)
    const float* V   = (const float*)d_in[6];  // (NBG, NN)
    float* out = (float*)d_out;                // [x_new | r_new], each NN*BATCH
    float* t   = (float*)d_ws;                 // NBG*BATCH floats = 128 KB

    thal_vr_gate_kernel<<<dim3(BATCH / 64), 256, 0, stream>>>(V, r, rth, t);

    dim3 g2(NN / 128, BATCH / 64);
    thal_rnn_step_kernel<<<g2, 256, 0, stream>>>(J, U, t, r, x, Bv, out);
}